// F_NSWT_28372553957439
// MI455X (gfx1250) — compile-verified
//
#include <hip/hip_runtime.h>
#include <math.h>

// ---------------- problem constants ----------------
#define HH   224
#define WWID 224
#define HWSZ 50176            // 224*224
#define CCH  64
#define C3CH 192
#define CHWSZ (64 * 50176)    // one [C,H,W] plane set

typedef __attribute__((ext_vector_type(2))) float v2f;
typedef __attribute__((ext_vector_type(8))) float v8f;

// fp32 WMMA 16x16x4: D = A(16x4) * B(4x16) + C, full fp32 precision
static __device__ inline v8f wmma_f32_4(v2f a, v2f b, v8f c) {
  return __builtin_amdgcn_wmma_f32_16x16x4_f32(false, a, false, b, (short)0, c, false, false);
}

// sym4 decomposition filters (module constants in the reference)
__constant__ float cDEC_LO[8] = {
  -0.07576571478927333f, -0.02963552764599851f, 0.49761866763201545f,
   0.8037387518059161f,   0.29785779560527736f, -0.09921954357684722f,
  -0.012603967262037833f, 0.0322231006040427f };
__constant__ float cDEC_HI[8] = {
  -0.0322231006040427f,  -0.012603967262037833f, 0.09921954357684722f,
   0.29785779560527736f, -0.8037387518059161f,   0.49761866763201545f,
   0.02963552764599851f, -0.07576571478927333f };
__constant__ int cMX[16] = {0,0,6,0,0,1,1,4,5,1,3,0,0,0,3,2};
__constant__ int cMY[16] = {0,1,0,5,2,0,2,0,0,6,0,4,6,3,5,2};

static __device__ inline float dct_b(int u, int j) {
  // basis(u,224)[j]; 1/sqrt(224) = 0.0668153104781...
  float v = cosf(3.14159265358979323846f * (float)u * ((float)j + 0.5f) / 224.0f)
            * 0.06681531047810609f;
  return (u != 0) ? v * 1.4142135623730951f : v;
}

// ---------------- SWT: one 8-tap circular a-trous pass ----------------
__global__ void swt_pass(const float* __restrict__ src, float* __restrict__ dst,
                         int dil, int axis, int hi) {
  int idx = blockIdx.x * blockDim.x + threadIdx.x;
  if (idx >= CHWSZ) return;
  int w = idx % WWID;
  int h = (idx / WWID) % HH;
  int c = idx / HWSZ;
  float s = 0.f;
  if (axis == 1) {
    int base = c * HWSZ + w;
    #pragma unroll
    for (int k = 0; k < 8; ++k) {
      int t = h + k * dil; if (t >= HH) t -= HH;          // k*dil <= 28
      float fk = hi ? cDEC_HI[k] : cDEC_LO[k];
      s += fk * src[base + t * WWID];
    }
  } else {
    int base = c * HWSZ + h * WWID;
    #pragma unroll
    for (int k = 0; k < 8; ++k) {
      int t = w + k * dil; if (t >= WWID) t -= WWID;
      float fk = hi ? cDEC_HI[k] : cDEC_LO[k];
      s += fk * src[base + t];
    }
  }
  dst[idx] = s;
}

// ---------- separable-DCT spectral pooling: P[i,ch] = sum bx[h] by[w] in[ch,h,w] ----------
__global__ void spectral_pool(const float* __restrict__ cA, const float* __restrict__ cHb,
                              const float* __restrict__ cVb, const float* __restrict__ cDb,
                              float* __restrict__ P, int nch, int mode) {
  int ch = blockIdx.x;   // channel
  int i  = blockIdx.y;   // frequency index 0..15
  int u = cMX[i] * 32, v = cMY[i] * 32;
  __shared__ float red[256];
  int t = threadIdx.x;
  float val = 0.f;
  if (t < WWID) {
    const float* src;
    if (mode == 0) {
      src = cA + (size_t)ch * HWSZ;
    } else {                        // high stack: c3 = 3*c + {H,V,D}
      int j = ch % 3, cc = ch / 3;
      src = ((j == 0) ? cHb : (j == 1) ? cVb : cDb) + (size_t)cc * HWSZ;
    }
    float inner = 0.f;
    for (int hrow = 0; hrow < HH; ++hrow)
      inner += dct_b(u, hrow) * src[hrow * WWID + t];
    val = inner * dct_b(v, t);
  }
  red[t] = val;
  __syncthreads();
  for (int sdiv = 128; sdiv > 0; sdiv >>= 1) {
    if (t < sdiv) red[t] += red[t + sdiv];
    __syncthreads();
  }
  if (t == 0) P[i * nch + ch] = red[0];
}

// ---------------- tiny attention MLPs -> sigmoid scales ----------------
__global__ void attn_scale(const float* __restrict__ P_low, const float* __restrict__ P_high,
                           const float* __restrict__ W_dl, const float* __restrict__ b_dl,
                           const float* __restrict__ W1_low, const float* __restrict__ W2_low,
                           const float* __restrict__ W_dh, const float* __restrict__ b_dh,
                           const float* __restrict__ W1_high, const float* __restrict__ W2_high,
                           float* __restrict__ scale_low, float* __restrict__ scale_high) {
  __shared__ float y_low[64], y_high[192], t_low[4], t_high[12];
  int t = threadIdx.x;
  if (t < 64) {
    int i = t >> 2;                           // cp = 64/16 = 4
    int u = cMX[i] * 32, v = cMY[i] * 32;
    float sbx = 0.f, sby = 0.f;
    for (int j = 0; j < 224; ++j) { sbx += dct_b(u, j); sby += dct_b(v, j); }
    float acc = b_dl[t] * sbx * sby;
    for (int c = 0; c < 64; ++c) acc += W_dl[t * 64 + c] * P_low[i * 64 + c];
    y_low[t] = acc;
  }
  if (t < 192) {
    int i = t / 12;                           // cp = 192/16 = 12
    int u = cMX[i] * 32, v = cMY[i] * 32;
    float sbx = 0.f, sby = 0.f;
    for (int j = 0; j < 224; ++j) { sbx += dct_b(u, j); sby += dct_b(v, j); }
    float acc = b_dh[t] * sbx * sby;
    for (int c = 0; c < 192; ++c) acc += W_dh[t * 192 + c] * P_high[i * 192 + c];
    y_high[t] = acc;
  }
  __syncthreads();
  if (t < 4) {
    float a = 0.f;
    for (int c = 0; c < 64; ++c) a += W1_low[t * 64 + c] * y_low[c];
    t_low[t] = a > 0.f ? a : 0.f;
  }
  if (t >= 32 && t < 44) {
    int j = t - 32; float a = 0.f;
    for (int c = 0; c < 192; ++c) a += W1_high[j * 192 + c] * y_high[c];
    t_high[j] = a > 0.f ? a : 0.f;
  }
  __syncthreads();
  if (t < 64) {
    float a = 0.f;
    for (int j = 0; j < 4; ++j) a += W2_low[t * 4 + j] * t_low[j];
    scale_low[t] = 1.f / (1.f + expf(-a));
  }
  if (t < 192) {
    float a = 0.f;
    for (int j = 0; j < 12; ++j) a += W2_high[t * 12 + j] * t_high[j];
    scale_high[t] = 1.f / (1.f + expf(-a));
  }
}

static __device__ inline float high_elem(const float* __restrict__ cHb,
                                         const float* __restrict__ cVb,
                                         const float* __restrict__ cDb, int c3, int p) {
  int j = c3 % 3, c = c3 / 3;
  const float* b = (j == 0) ? cHb : (j == 1) ? cVb : cDb;
  return b[(size_t)c * HWSZ + p];
}

// --------- 1x1 attention convs via WMMA: dmt[0:64]=low_att, dmt[64:256]=high_att ---------
__global__ __launch_bounds__(256) void dmt_build(
    const float* __restrict__ cA, const float* __restrict__ cHb,
    const float* __restrict__ cVb, const float* __restrict__ cDb,
    const float* __restrict__ W_al, const float* __restrict__ b_al,
    const float* __restrict__ W_ah, const float* __restrict__ b_ah,
    const float* __restrict__ scale_low, const float* __restrict__ scale_high,
    float* __restrict__ dmt) {
  int lane = threadIdx.x & 31, wave = threadIdx.x >> 5;
  int idx15 = lane & 15, khf = lane >> 4;
  int tile = blockIdx.x * 8 + wave;         // 3136 pixel tiles of 16
  int p0 = tile * 16;
  int ot = blockIdx.y;                      // 0..3 low octiles, 4..15 high octiles
  v8f acc = {0.f,0.f,0.f,0.f,0.f,0.f,0.f,0.f};
  if (ot < 4) {
    int ob = ot * 16;
    const float* brow = W_al + (size_t)(ob + idx15) * 64;   // row is 8B aligned
    for (int kc = 0; kc < 16; ++kc) {       // K = 64 channels
      int cb = kc * 4 + 2 * khf;
      v2f a;
      const float* ap = cA + (size_t)cb * HWSZ + p0 + idx15;
      a.x = ap[0]; a.y = ap[HWSZ];
      v2f b = *(const v2f*)(brow + cb);     // cb even -> 8B aligned -> b64 load
      acc = wmma_f32_4(a, b, acc);
    }
    int oc = ob + idx15;
    float bias = b_al[oc], sc = scale_low[oc];
    float* op = dmt + (size_t)oc * HWSZ + p0 + 8 * khf;
    #pragma unroll
    for (int r = 0; r < 8; ++r) op[r] = (acc[r] + bias) * sc;
  } else {
    int ob = (ot - 4) * 16;
    const float* brow = W_ah + (size_t)(ob + idx15) * 192;  // row is 8B aligned
    for (int kc = 0; kc < 48; ++kc) {       // K = 192 channels
      int c3 = kc * 4 + 2 * khf;
      v2f a;
      a.x = high_elem(cHb, cVb, cDb, c3,     p0 + idx15);
      a.y = high_elem(cHb, cVb, cDb, c3 + 1, p0 + idx15);
      v2f b = *(const v2f*)(brow + c3);     // c3 even -> 8B aligned
      acc = wmma_f32_4(a, b, acc);
    }
    int oc = ob + idx15;
    float bias = b_ah[oc], sc = scale_high[oc];
    float* op = dmt + (size_t)(64 + oc) * HWSZ + p0 + 8 * khf;
    #pragma unroll
    for (int r = 0; r < 8; ++r) op[r] = (acc[r] + bias) * sc;
  }
}

// ---------- conv3x3 256->64 (batch 1) + leaky relu, implicit GEMM w/ LDS-staged weights ----------
// Ws2 layout: [K-pair pp][oc][2]  -> B operand per lane is one aligned 8B ds_load_b64.
__global__ __launch_bounds__(256) void enc_conv(const float* __restrict__ dmt,
                                                const float* __restrict__ W_enc,
                                                const float* __restrict__ b_enc,
                                                float* __restrict__ enc) {
  __shared__ float Ws2[64 * 64];            // 32 K-pairs x 64 oc x 2, 16 KB
  int lane = threadIdx.x & 31, wave = threadIdx.x >> 5;
  int idx15 = lane & 15, khf = lane >> 4;
  int tile = blockIdx.x * 8 + wave;         // 16 px per wave; 224%16==0 -> tile never crosses a row
  int p0 = tile * 16, h = p0 / WWID, w0 = p0 - h * WWID;
  v8f acc[4];
  #pragma unroll
  for (int nt = 0; nt < 4; ++nt) acc[nt] = (v8f){0.f,0.f,0.f,0.f,0.f,0.f,0.f,0.f};
  for (int ky = 0; ky < 3; ++ky) {
    int hh = h + ky - 1;
    bool okh = (hh >= 0) && (hh < HH);
    int hc = hh < 0 ? 0 : (hh > HH - 1 ? HH - 1 : hh);
    for (int kx = 0; kx < 3; ++kx) {
      int ww = w0 + idx15 + kx - 1;
      bool ok = okh && (ww >= 0) && (ww < WWID);
      int wc = ww < 0 ? 0 : (ww > WWID - 1 ? WWID - 1 : ww);
      float msk = ok ? 1.f : 0.f;
      for (int cc = 0; cc < 4; ++cc) {      // 4 chunks of 64 input channels
        __syncthreads();
        for (int t2 = threadIdx.x; t2 < 4096; t2 += 256) {
          int pp = t2 >> 7, oc = (t2 >> 1) & 63, half = t2 & 1;
          int ic = cc * 64 + pp * 2 + half;
          Ws2[t2] = W_enc[((oc * 256 + ic) * 3 + ky) * 3 + kx];
        }
        __syncthreads();
        // speculative prefetch of next chunk's weight slice -> global_prefetch_b8
        if (cc < 3) {
          int oc = threadIdx.x & 63, icn = (cc + 1) * 64 + (threadIdx.x >> 6) * 16;
          __builtin_prefetch(&W_enc[((oc * 256 + icn) * 3 + ky) * 3 + kx], 0, 0);
        }
        const float* ap0 = dmt + (size_t)(cc * 64 + 2 * khf) * HWSZ + hc * WWID + wc;
        for (int kc = 0; kc < 16; ++kc) {
          v2f a;
          const float* ap = ap0 + (size_t)(kc * 4) * HWSZ;
          a.x = msk * ap[0]; a.y = msk * ap[HWSZ];
          int pp = kc * 2 + khf;            // K-pair index (icl = kc*4 + 2*khf)
          #pragma unroll
          for (int nt = 0; nt < 4; ++nt) {
            v2f b = *(const v2f*)&Ws2[pp * 128 + (nt * 16 + idx15) * 2]; // ds_load_b64
            acc[nt] = wmma_f32_4(a, b, acc[nt]);
          }
        }
      }
    }
  }
  #pragma unroll
  for (int nt = 0; nt < 4; ++nt) {
    int oc = nt * 16 + idx15;
    float bias = b_enc[oc];
    float* op = enc + (size_t)oc * HWSZ + p0 + 8 * khf;
    #pragma unroll
    for (int r = 0; r < 8; ++r) {
      float vv = acc[nt][r] + bias;
      op[r] = vv > 0.f ? vv : 0.01f * vv;   // leaky relu
    }
  }
}

// ---------- conv3x3 64->64 for all 8 batches, + bias + enc broadcast add ----------
__global__ __launch_bounds__(256) void res_conv(const float* __restrict__ x,
                                                const float* __restrict__ W_id,
                                                const float* __restrict__ b_id,
                                                const float* __restrict__ enc,
                                                float* __restrict__ out) {
  __shared__ float Ws2[64 * 64];            // pair-interleaved, 16 KB
  int lane = threadIdx.x & 31, wave = threadIdx.x >> 5;
  int idx15 = lane & 15, khf = lane >> 4;
  int bi = blockIdx.y;
  int tile = blockIdx.x * 8 + wave;
  int p0 = tile * 16, h = p0 / WWID, w0 = p0 - h * WWID;
  const float* xb = x + (size_t)bi * CHWSZ;
  v8f acc[4];
  #pragma unroll
  for (int nt = 0; nt < 4; ++nt) acc[nt] = (v8f){0.f,0.f,0.f,0.f,0.f,0.f,0.f,0.f};
  for (int ky = 0; ky < 3; ++ky) {
    int hh = h + ky - 1;
    bool okh = (hh >= 0) && (hh < HH);
    int hc = hh < 0 ? 0 : (hh > HH - 1 ? HH - 1 : hh);
    for (int kx = 0; kx < 3; ++kx) {
      __syncthreads();
      for (int t2 = threadIdx.x; t2 < 4096; t2 += 256) {
        int pp = t2 >> 7, oc = (t2 >> 1) & 63, half = t2 & 1;
        int ic = pp * 2 + half;
        Ws2[t2] = W_id[((oc * 64 + ic) * 3 + ky) * 3 + kx];
      }
      __syncthreads();
      // prefetch next (ky,kx) weight slice
      if (!(ky == 2 && kx == 2)) {
        int kxn = (kx + 1) % 3, kyn = ky + (kx == 2 ? 1 : 0);
        int oc = threadIdx.x & 63, icn = (threadIdx.x >> 6) * 16;
        __builtin_prefetch(&W_id[((oc * 64 + icn) * 3 + kyn) * 3 + kxn], 0, 0);
      }
      int ww = w0 + idx15 + kx - 1;
      bool ok = okh && (ww >= 0) && (ww < WWID);
      int wc = ww < 0 ? 0 : (ww > WWID - 1 ? WWID - 1 : ww);
      float msk = ok ? 1.f : 0.f;
      const float* ap0 = xb + (size_t)(2 * khf) * HWSZ + hc * WWID + wc;
      for (int kc = 0; kc < 16; ++kc) {
        v2f a;
        const float* ap = ap0 + (size_t)(kc * 4) * HWSZ;
        a.x = msk * ap[0]; a.y = msk * ap[HWSZ];
        int pp = kc * 2 + khf;
        #pragma unroll
        for (int nt = 0; nt < 4; ++nt) {
          v2f b = *(const v2f*)&Ws2[pp * 128 + (nt * 16 + idx15) * 2]; // ds_load_b64
          acc[nt] = wmma_f32_4(a, b, acc[nt]);
        }
      }
    }
  }
  #pragma unroll
  for (int nt = 0; nt < 4; ++nt) {
    int oc = nt * 16 + idx15;
    float bias = b_id[oc];
    const float* ep = enc + (size_t)oc * HWSZ + p0 + 8 * khf;
    float* op = out + ((size_t)(bi * 64 + oc)) * HWSZ + p0 + 8 * khf;
    #pragma unroll
    for (int r = 0; r < 8; ++r) op[r] = acc[nt][r] + bias + ep[r];
  }
}

// ---------------- host-side orchestration ----------------
extern "C" void kernel_launch(void* const* d_in, const int* in_sizes, int n_in,
                              void* d_out, int out_size, void* d_ws, size_t ws_size,
                              hipStream_t stream) {
  (void)in_sizes; (void)n_in; (void)out_size; (void)ws_size;
  const float* x      = (const float*)d_in[0];
  const float* W_id   = (const float*)d_in[1];
  const float* b_id   = (const float*)d_in[2];
  const float* W_dl   = (const float*)d_in[3];
  const float* b_dl   = (const float*)d_in[4];
  const float* W_al   = (const float*)d_in[5];
  const float* b_al   = (const float*)d_in[6];
  const float* W1_low = (const float*)d_in[7];
  const float* W2_low = (const float*)d_in[8];
  const float* W_dh   = (const float*)d_in[9];
  const float* b_dh   = (const float*)d_in[10];
  const float* W_ah   = (const float*)d_in[11];
  const float* b_ah   = (const float*)d_in[12];
  const float* W1_high= (const float*)d_in[13];
  const float* W2_high= (const float*)d_in[14];
  const float* W_enc  = (const float*)d_in[15];
  const float* b_enc  = (const float*)d_in[16];
  float* out = (float*)d_out;

  // workspace layout (floats): tmp|abuf|cA|cH|cV|cD|dmt(4 planes)|enc|small
  float* ws   = (float*)d_ws;
  float* tmp  = ws;
  float* abuf = ws + 1 * (size_t)CHWSZ;
  float* cAb  = ws + 2 * (size_t)CHWSZ;
  float* cHb  = ws + 3 * (size_t)CHWSZ;
  float* cVb  = ws + 4 * (size_t)CHWSZ;
  float* cDb  = ws + 5 * (size_t)CHWSZ;
  float* dmt  = ws + 6 * (size_t)CHWSZ;    // 256 channels
  float* encb = ws + 10 * (size_t)CHWSZ;
  float* P_low  = ws + 11 * (size_t)CHWSZ;
  float* P_high = P_low + 16 * 64;
  float* s_low  = P_high + 16 * 192;
  float* s_high = s_low + 64;
  float* aHtmp  = dmt;                     // dmt region free during SWT

  dim3 eb(256);
  dim3 eg(CHWSZ / 256);                    // 12544, exact
  // levels 1 & 2: lowpass rows then cols (dil 1, 2)
  swt_pass<<<eg, eb, 0, stream>>>(x,    tmp,  1, 1, 0);
  swt_pass<<<eg, eb, 0, stream>>>(tmp,  abuf, 1, 2, 0);
  swt_pass<<<eg, eb, 0, stream>>>(abuf, tmp,  2, 1, 0);
  swt_pass<<<eg, eb, 0, stream>>>(tmp,  abuf, 2, 2, 0);
  // level 3 (dil 4): aL -> tmp, aH -> aHtmp, then four column filters
  swt_pass<<<eg, eb, 0, stream>>>(abuf, tmp,   4, 1, 0);
  swt_pass<<<eg, eb, 0, stream>>>(abuf, aHtmp, 4, 1, 1);
  swt_pass<<<eg, eb, 0, stream>>>(tmp,   cAb, 4, 2, 0);
  swt_pass<<<eg, eb, 0, stream>>>(tmp,   cHb, 4, 2, 1);
  swt_pass<<<eg, eb, 0, stream>>>(aHtmp, cVb, 4, 2, 0);
  swt_pass<<<eg, eb, 0, stream>>>(aHtmp, cDb, 4, 2, 1);

  // separable spectral pooling (replaces full DCT-filter weighted sums)
  spectral_pool<<<dim3(64, 16),  256, 0, stream>>>(cAb, cHb, cVb, cDb, P_low, 64, 0);
  spectral_pool<<<dim3(192, 16), 256, 0, stream>>>(cAb, cHb, cVb, cDb, P_high, 192, 1);

  attn_scale<<<1, 256, 0, stream>>>(P_low, P_high, W_dl, b_dl, W1_low, W2_low,
                                    W_dh, b_dh, W1_high, W2_high, s_low, s_high);

  // 1x1 attention convs (WMMA) -> dmt [256,H,W]
  dmt_build<<<dim3(392, 16), 256, 0, stream>>>(cAb, cHb, cVb, cDb,
                                               W_al, b_al, W_ah, b_ah,
                                               s_low, s_high, dmt);

  // 3x3 conv 256->64 + leaky relu (WMMA implicit GEMM)
  enc_conv<<<dim3(392), 256, 0, stream>>>(dmt, W_enc, b_enc, encb);

  // 3x3 conv 64->64 for all 8 batches + bias + enc broadcast (WMMA implicit GEMM)
  res_conv<<<dim3(392, 8), 256, 0, stream>>>(x, W_id, b_id, encb, out);
}